// MHAJAM_67534065762578
// MI455X (gfx1250) — compile-verified
//
#include <hip/hip_runtime.h>
#include <stdint.h>

// Problem constants (from the reference)
#define B_    64
#define N_SV  63
#define HID   512
#define NW    28
#define NH    28
#define CELLS (NW * NH)          // 784
#define HT    8                  // h-values handled per block
#define NTH   256                // 8 wave32 per block
#define PLANE (HT * CELLS)       // 6272 floats = 25088 bytes of LDS
#define VEC4  (PLANE / 4)        // 1568 x 16B chunks
#define FULL  (VEC4 / NTH)       // 6 full rounds
#define TAIL  (VEC4 - FULL*NTH)  // 32 tail chunks

__global__ __launch_bounds__(NTH)
void MHAJAM_scatter_max_kernel(const float* __restrict__ pos,   // (B, N, 3)
                               const float* __restrict__ enc,   // (B, N, HID)
                               const int*   __restrict__ lengths,
                               float*       __restrict__ out)   // (B, HID, NW, NH)
{
    __shared__ float plane[PLANE];   // [hh][x*NH + y], matches global layout

    const int tid = threadIdx.x;
    const int b   = blockIdx.x >> 6;          // HID/HT == 64 h-blocks per batch
    const int h0  = (blockIdx.x & 63) * HT;

    // ---- Phase 1: zero the LDS tile (ds_store_b128, fully unrolled) ----
    float4* p4 = (float4*)plane;
    const float4 z4 = make_float4(0.f, 0.f, 0.f, 0.f);
#pragma unroll
    for (int k = 0; k < FULL; ++k)
        p4[k * NTH + tid] = z4;
    if (tid < TAIL)
        p4[FULL * NTH + tid] = z4;
    __syncthreads();

    const int len = lengths[b];   // uniform scalar load

    // ---- Phase 2: scatter-max agents into LDS via integer ds_max_u32 ----
    // Grid starts at 0 and only max-updates => every cell is >= 0 forever,
    // so writes with value <= 0 (including all masked agents) are no-ops.
    // Remaining effective values are strictly positive floats, whose IEEE
    // bit patterns are monotone as u32 => native ds_max_u32 suffices.
#pragma unroll
    for (int it = 0; it < 2; ++it) {
        const int idx = it * NTH + tid;
        if (idx < N_SV * HT) {
            const int n  = idx >> 3;          // HT == 8
            const int hh = idx & (HT - 1);

            const float x = pos[(b * N_SV + n) * 3 + 0];
            const float y = pos[(b * N_SV + n) * 3 + 1];
            // Reference op order exactly: (x * NEW_W) / OLD_W, trunc toward 0
            const int xIdx = (int)((x * 28.0f) / 224.0f);
            const int yIdx = (int)((y * 28.0f) / 224.0f);

            if ((n < len) && (xIdx < NW) && (yIdx < NH)) {
                const int xc = min(max(xIdx, 0), NW - 1);
                const int yc = min(max(yIdx, 0), NH - 1);
                const float v = enc[((size_t)(b * N_SV + n)) * HID + h0 + hh];
                if (v > 0.0f) {
                    unsigned* cellp =
                        (unsigned*)&plane[hh * CELLS + xc * NH + yc];
                    atomicMax(cellp, __float_as_uint(v));   // ds_max_u32
                }
            }
        }
    }
    __syncthreads();

    // ---- Phase 3: stream LDS tile -> global with the CDNA5 async engine ----
    // Destination slab out[b, h0:h0+HT, :, :] is one contiguous 25088B range.
    const uint64_t gbase =
        (uint64_t)(uintptr_t)(out + ((size_t)b * HID + h0) * CELLS);
    // Low 32 bits of the flat shared-aperture address == LDS byte offset.
    const uint32_t lbase = (uint32_t)(uintptr_t)&plane[0];

#pragma unroll
    for (int k = 0; k < FULL; ++k) {
        const uint32_t voff = (uint32_t)((k * NTH + tid) * 16);
        const uint32_t lsrc = lbase + voff;
        // GVS mode: mem_addr = SGPR64 + VGPR32 ; LDS source addr from VGPR.
        asm volatile("global_store_async_from_lds_b128 %0, %1, %2"
                     :
                     : "v"(voff), "v"(lsrc), "s"(gbase)
                     : "memory");
    }
    if (tid < TAIL) {
        const uint32_t voff = (uint32_t)((FULL * NTH + tid) * 16);
        const uint32_t lsrc = lbase + voff;
        asm volatile("global_store_async_from_lds_b128 %0, %1, %2"
                     :
                     : "v"(voff), "v"(lsrc), "s"(gbase)
                     : "memory");
    }
    // Drain ASYNCcnt before the wave retires.
    asm volatile("s_wait_asynccnt 0" ::: "memory");
}

extern "C" void kernel_launch(void* const* d_in, const int* in_sizes, int n_in,
                              void* d_out, int out_size, void* d_ws, size_t ws_size,
                              hipStream_t stream) {
    const float* pos     = (const float*)d_in[0];   // svPositionsAtT0 (B,N,3)
    const float* enc     = (const float*)d_in[1];   // svEncoding      (B,N,HID)
    const int*   lengths = (const int*)d_in[2];     // (B,)
    float*       out     = (float*)d_out;           // (B,HID,NW,NH)

    const dim3 grid(B_ * (HID / HT));  // 64 * 64 = 4096 blocks
    const dim3 block(NTH);
    MHAJAM_scatter_max_kernel<<<grid, block, 0, stream>>>(pos, enc, lengths, out);
}